// DataAugmentation_41970420418237
// MI455X (gfx1250) — compile-verified
//
#include <hip/hip_runtime.h>
#include <math.h>

// Problem constants (B,1,512,512 images; 2-channel noise; 85-tap Gaussian)
#define NB   64
#define HH   512
#define WW   512
#define HWSZ (HH * WW)
#define KS   85
#define PADR 42          // KS/2
#define SIGG 24.0f
#define ALPH 2.0f

typedef __attribute__((ext_vector_type(2))) float v2f;
typedef __attribute__((ext_vector_type(8))) float v8f;

__device__ __forceinline__ int refl512(int i) {
    // jnp.pad 'reflect' (no edge repeat); pad=42 < 512 so one reflection suffices
    i = (i < 0) ? -i : i;
    i = (i > HH - 1) ? (2 * (HH - 1) - i) : i;
    return i;
}

// ---------------------------------------------------------------------------
// Kernel 0: normalized Gaussian weights -> ws
// ---------------------------------------------------------------------------
__global__ void __launch_bounds__(128) gw_kernel(float* __restrict__ w) {
    __shared__ float g[128];
    __shared__ float s;
    int i = threadIdx.x;
    float v = 0.0f;
    if (i < KS) {
        float xv = (float)i - (float)(KS - 1) * 0.5f;
        float t = xv / SIGG;
        v = expf(-0.5f * t * t);
    }
    g[i] = v;
    __syncthreads();
    if (i == 0) {
        float acc = 0.0f;
        for (int j = 0; j < KS; ++j) acc += g[j];
        s = acc;
    }
    __syncthreads();
    if (i < KS) w[i] = g[i] / s;
}

// ---------------------------------------------------------------------------
// Kernel 1: vertical blur of (2*noise - 1) via V_WMMA_F32_16X16X4_F32.
// One wave computes a 16x16 tile: D[y,x] = sum_k A[y,k] * B[k,x], K = 100.
//   A[y,k] = w[k - y]              (banded Toeplitz, zero-padded LDS table)
//   B[k,x] = 2*noise[refl(row),c]-1
// A/B lane layout (ISA 7.12.2): M/N = lane&15, K = 2*(lane>>4) + vgpr.
// C/D layout: row M = r + 8*(lane>>4), col N = lane&15.
// ---------------------------------------------------------------------------
__global__ void __launch_bounds__(256) blurV_wmma(const float* __restrict__ noise,
                                                  float* __restrict__ tmp,
                                                  const float* __restrict__ w) {
    __shared__ float lw[128];   // lw[u+16] = w[u] for u in [0,85), else 0
    int t = threadIdx.x;
    if (t < 128) lw[t] = (t >= 16 && t < 16 + KS) ? w[t - 16] : 0.0f;
    __syncthreads();

    int lane = t & 31;
    int wv   = t >> 5;
    int tile = blockIdx.x * 8 + wv;      // grid sized exactly: no guard, EXEC all-1s
    int img  = tile >> 10;               // 1024 tiles per (b,c) image
    int tt   = tile & 1023;
    int ybase = (tt >> 5) << 4;
    int xbase = (tt & 31) << 4;

    const float* src = noise + (size_t)img * HWSZ;
    float*       dst = tmp   + (size_t)img * HWSZ;

    int half = lane >> 4;
    int lm   = lane & 15;
    int col  = xbase + lm;

    v8f acc = {};
    for (int c = 0; c < 25; ++c) {       // 25 chunks of K=4 -> K=100
        int kk = 4 * c + 2 * half;
        v2f a, bb;
        int u = kk - lm + 16;            // in [1,116) subset of [0,128)
        a.x = lw[u];
        a.y = lw[u + 1];
        int r0 = refl512(ybase - PADR + kk);
        int r1 = refl512(ybase - PADR + kk + 1);
        bb.x = 2.0f * src[r0 * WW + col] - 1.0f;
        bb.y = 2.0f * src[r1 * WW + col] - 1.0f;
        acc = __builtin_amdgcn_wmma_f32_16x16x4_f32(false, a, false, bb,
                                                    (short)0, acc, false, false);
    }
    int mrow = ybase + 8 * half;
#pragma unroll
    for (int r = 0; r < 8; ++r)
        dst[(mrow + r) * WW + col] = acc[r];
}

// ---------------------------------------------------------------------------
// Kernel 2: horizontal blur of tmp, output disp = ALPHA * blurred.
//   D[y,x] = sum_k A[y,k] * B[k,x];  A[y,k] = tmp[y, refl(xbase-42+k)],
//   B[k,x] = w[k - x].
// ---------------------------------------------------------------------------
__global__ void __launch_bounds__(256) blurH_wmma(const float* __restrict__ tmp,
                                                  float* __restrict__ disp,
                                                  const float* __restrict__ w) {
    __shared__ float lw[128];
    int t = threadIdx.x;
    if (t < 128) lw[t] = (t >= 16 && t < 16 + KS) ? w[t - 16] : 0.0f;
    __syncthreads();

    int lane = t & 31;
    int wv   = t >> 5;
    int tile = blockIdx.x * 8 + wv;
    int img  = tile >> 10;
    int tt   = tile & 1023;
    int ybase = (tt >> 5) << 4;
    int xbase = (tt & 31) << 4;

    const float* src = tmp  + (size_t)img * HWSZ;
    float*       dst = disp + (size_t)img * HWSZ;

    int half = lane >> 4;
    int lm   = lane & 15;

    v8f acc = {};
    for (int c = 0; c < 25; ++c) {
        int kk = 4 * c + 2 * half;
        v2f a, bb;
        int c0 = refl512(xbase - PADR + kk);
        int c1 = refl512(xbase - PADR + kk + 1);
        a.x = src[(ybase + lm) * WW + c0];
        a.y = src[(ybase + lm) * WW + c1];
        int u = kk - lm + 16;
        bb.x = lw[u];
        bb.y = lw[u + 1];
        acc = __builtin_amdgcn_wmma_f32_16x16x4_f32(false, a, false, bb,
                                                    (short)0, acc, false, false);
    }
    int mrow = ybase + 8 * half;
    int col  = xbase + lm;
#pragma unroll
    for (int r = 0; r < 8; ++r)
        dst[(mrow + r) * WW + col] = ALPH * acc[r];
}

// ---------------------------------------------------------------------------
// Kernel 3: bilinear grid-sample of img and seg with zero-padding masks.
// out[0..N)   = warped img (f32)
// out[N..2N)  = trunc(warped seg)  (reference: astype int64, stored as f32)
// ---------------------------------------------------------------------------
__global__ void __launch_bounds__(256) warp_kernel(const float* __restrict__ img,
                                                   const float* __restrict__ seg,
                                                   const float* __restrict__ disp,
                                                   float* __restrict__ out) {
    size_t N   = (size_t)NB * HWSZ;
    size_t idx = (size_t)blockIdx.x * blockDim.x + threadIdx.x;
    if (idx >= N) return;

    int b   = (int)(idx / HWSZ);
    int rem = (int)(idx % HWSZ);
    int y   = rem >> 9;          // /512
    int x   = rem & 511;

    float dx = disp[((size_t)b * 2 + 0) * HWSZ + rem];
    float dy = disp[((size_t)b * 2 + 1) * HWSZ + rem];

    float gx = -1.0f + 2.0f * (float)x / (float)(WW - 1) + dx;
    float gy = -1.0f + 2.0f * (float)y / (float)(HH - 1) + dy;
    gx = fminf(fmaxf(gx, -1.0f), 1.0f);
    gy = fminf(fmaxf(gy, -1.0f), 1.0f);

    float px = ((gx + 1.0f) * (float)WW - 1.0f) * 0.5f;
    float py = ((gy + 1.0f) * (float)HH - 1.0f) * 0.5f;

    float x0f = floorf(px), y0f = floorf(py);
    int   x0  = (int)x0f,   y0  = (int)y0f;
    int   x1  = x0 + 1,     y1  = y0 + 1;
    float wx1 = px - x0f,   wy1 = py - y0f;
    float wx0 = 1.0f - wx1, wy0 = 1.0f - wy1;

    float v00 = (x0 >= 0 && x0 < WW && y0 >= 0 && y0 < HH) ? 1.0f : 0.0f;
    float v10 = (x1 >= 0 && x1 < WW && y0 >= 0 && y0 < HH) ? 1.0f : 0.0f;
    float v01 = (x0 >= 0 && x0 < WW && y1 >= 0 && y1 < HH) ? 1.0f : 0.0f;
    float v11 = (x1 >= 0 && x1 < WW && y1 >= 0 && y1 < HH) ? 1.0f : 0.0f;

    int x0c = min(max(x0, 0), WW - 1), x1c = min(max(x1, 0), WW - 1);
    int y0c = min(max(y0, 0), HH - 1), y1c = min(max(y1, 0), HH - 1);

    float w00 = wx0 * wy0 * v00, w10 = wx1 * wy0 * v10;
    float w01 = wx0 * wy1 * v01, w11 = wx1 * wy1 * v11;

    const float* ib = img + (size_t)b * HWSZ;
    const float* sb = seg + (size_t)b * HWSZ;
    int i00 = y0c * WW + x0c, i10 = y0c * WW + x1c;
    int i01 = y1c * WW + x0c, i11 = y1c * WW + x1c;

    float vi = ib[i00] * w00 + ib[i10] * w10 + ib[i01] * w01 + ib[i11] * w11;
    float vs = sb[i00] * w00 + sb[i10] * w10 + sb[i01] * w01 + sb[i11] * w11;

    out[idx]     = vi;
    out[N + idx] = (float)(long long)vs;   // astype(int64): truncation toward zero
}

// ---------------------------------------------------------------------------
extern "C" void kernel_launch(void* const* d_in, const int* in_sizes, int n_in,
                              void* d_out, int out_size, void* d_ws, size_t ws_size,
                              hipStream_t stream) {
    (void)in_sizes; (void)n_in; (void)out_size; (void)ws_size;
    const float* img   = (const float*)d_in[0];
    const float* seg   = (const float*)d_in[1];
    const float* noise = (const float*)d_in[2];
    float*       out   = (float*)d_out;

    // Workspace layout: [128 f32 weights][tmp: 64*2*512*512 f32][disp: same]
    float* wgt  = (float*)d_ws;
    float* tmp  = wgt + 128;
    float* disp = tmp + (size_t)NB * 2 * HWSZ;

    gw_kernel<<<1, 128, 0, stream>>>(wgt);

    // 64*2 images * (32*32) tiles = 131072 wave-tiles, 8 waves/block
    blurV_wmma<<<131072 / 8, 256, 0, stream>>>(noise, tmp, wgt);
    blurH_wmma<<<131072 / 8, 256, 0, stream>>>(tmp, disp, wgt);

    size_t N = (size_t)NB * HWSZ;
    warp_kernel<<<(unsigned)((N + 255) / 256), 256, 0, stream>>>(img, seg, disp, out);
}